// GroupPooling_77068893159761
// MI455X (gfx1250) — compile-verified
//
#include <hip/hip_runtime.h>

#define HIDDEN 256
#define NAG    400000
#define NTM    100000

typedef float v2f __attribute__((ext_vector_type(2)));
typedef float v8f __attribute__((ext_vector_type(8)));
typedef unsigned int v4u __attribute__((ext_vector_type(4)));
typedef int v4i __attribute__((ext_vector_type(4)));
typedef int v8i __attribute__((ext_vector_type(8)));

// CDNA5 hardware tanh (TRANS op). v_nop covers the trans-result-use hazard.
__device__ __forceinline__ float tanh_hw(float x) {
  float y;
  asm volatile("v_tanh_f32 %0, %1\n\tv_nop" : "=v"(y) : "v"(x));
  return y;
}

// Tensor Data Mover: DMA `nelem` contiguous f32 from global into LDS.
// Descriptor per CDNA5 ISA: 2D tile 16384 x rows,
// data_size=4B, stride=16384, type=2 ("image"), count=1 user descriptor.
__device__ __forceinline__ void tdm_load_f32(const float* gsrc, float* ldst,
                                             unsigned nelem) {
  const unsigned long long ga = (unsigned long long)(const void*)gsrc;
  const unsigned lo = (unsigned)(unsigned long long)(void*)ldst; // LDS offset
  const unsigned d0 = 16384u;
  const unsigned rows = nelem / d0;

  v4u g0;
  g0.x = 1u;                                        // count=1 (valid user D#)
  g0.y = lo;                                        // lds_addr
  g0.z = (unsigned)ga;                              // global_addr[31:0]
  g0.w = (unsigned)((ga >> 32) & 0x01FFFFFFu) | (2u << 30); // addr[56:32]|type=2

  v8i g1;
  g1[0] = 0x00020000;                 // data_size=2 -> 4 bytes; no multicast
  g1[1] = (int)(d0 << 16);            // tensor_dim0[15:0]  @ bits 63:48
  g1[2] = (int)((d0 >> 16) | (rows << 16)); // tensor_dim0[31:16], tensor_dim1[15:0]
  g1[3] = (int)((rows >> 16) | (d0 << 16)); // tensor_dim1[31:16], tile_dim0
  g1[4] = (int)rows;                  // tile_dim1 (tile_dim2=0 unused)
  g1[5] = (int)d0;                    // tensor_dim0_stride[31:0]
  g1[6] = (int)(d0 << 16);            // stride0[47:32]=0, tensor_dim1_stride[15:0]
  g1[7] = 0;                          // tensor_dim1_stride[47:16]

  v4i gz4 = {0, 0, 0, 0};             // groups 2/3: tile_dim3/4 = 0 (unused)
  v8i gz8 = {0, 0, 0, 0, 0, 0, 0, 0}; // extra group (6-arg toolchain form)
  __builtin_amdgcn_tensor_load_to_lds(g0, g1, gz4, gz4, gz8, 0);
}

__global__ void zero_kernel(int* __restrict__ p, int n) {
  int i = blockIdx.x * 256 + threadIdx.x;
  if (i < n) p[i] = 0;
}

__global__ void count_kernel(const int* __restrict__ team_idx, int* __restrict__ counts) {
  int i = blockIdx.x * 256 + threadIdx.x;
  if (i < NAG) atomicAdd(&counts[team_idx[i]], 1);
}

// Single-workgroup looped exclusive scan over NTM counts -> offsets[NTM+1], cursor copy.
__global__ void scan_kernel(const int* __restrict__ counts,
                            int* __restrict__ offsets, int* __restrict__ cursor) {
  __shared__ int sh[1024];
  __shared__ int carry_sh;
  const int tid = threadIdx.x;
  if (tid == 0) carry_sh = 0;
  __syncthreads();
  for (int base = 0; base < NTM; base += 1024) {
    const int i = base + tid;
    const int v = (i < NTM) ? counts[i] : 0;
    sh[tid] = v;
    __syncthreads();
    for (int off = 1; off < 1024; off <<= 1) {
      int add = (tid >= off) ? sh[tid - off] : 0;
      __syncthreads();
      sh[tid] += add;
      __syncthreads();
    }
    const int incl  = sh[tid];
    const int carry = carry_sh;
    __syncthreads();
    if (i < NTM) {
      const int e = carry + incl - v;
      offsets[i] = e;
      cursor[i]  = e;
    }
    if (tid == 1023) carry_sh = carry + incl;
    __syncthreads();
  }
  if (tid == 0) offsets[NTM] = carry_sh;
}

__global__ void scatter_kernel(const int* __restrict__ team_idx,
                               int* __restrict__ cursor, int* __restrict__ bucket) {
  int i = blockIdx.x * 256 + threadIdx.x;
  if (i < NAG) {
    int p = atomicAdd(&cursor[team_idx[i]], 1);
    bucket[p] = i;
  }
}

// scores = tanh(agent_h @ W1 + b1) @ W2 + b2, one wave per 16 agents.
// W1 (128 KB) is DMA'd into LDS once per workgroup by the TDM; the fp32 WMMA
// inner loop then feeds B from LDS (ds_load), A from registers.
__global__ __launch_bounds__(256) void score_kernel(
    const float* __restrict__ agent_h, const float* __restrict__ W1,
    const float* __restrict__ b1, const float* __restrict__ W2,
    const float* __restrict__ b2, float* __restrict__ scores) {
  __shared__ float sW1[HIDDEN * 128];   // 32768 f32 = 128 KB

  if ((threadIdx.x >> 5) == 0) {        // wave 0 issues the tensor DMA
    tdm_load_f32(W1, sW1, HIDDEN * 128);
    __builtin_amdgcn_s_wait_tensorcnt(0);
  }
  __syncthreads();

  const int wave = blockIdx.x * 8 + (threadIdx.x >> 5);
  const int lane = threadIdx.x & 31;
  const int half = lane >> 4;
  const int lm   = lane & 15;

  const float* arow = agent_h + (size_t)(wave * 16 + lm) * HIDDEN + half * 2;
  v2f a[64];
#pragma unroll
  for (int kk = 0; kk < 64; ++kk)
    a[kk] = *(const v2f*)(arow + kk * 4);

  float sacc[8] = {0.f, 0.f, 0.f, 0.f, 0.f, 0.f, 0.f, 0.f};
  for (int j = 0; j < 8; ++j) {
    v8f c = {0.f, 0.f, 0.f, 0.f, 0.f, 0.f, 0.f, 0.f};
    const float* bcol = sW1 + (half * 2) * 128 + j * 16 + lm;
#pragma unroll
    for (int kk = 0; kk < 64; ++kk) {
      v2f b;
      b.x = bcol[kk * 4 * 128];
      b.y = bcol[kk * 4 * 128 + 128];
      c = __builtin_amdgcn_wmma_f32_16x16x4_f32(false, a[kk], false, b,
                                                (short)0, c, false, false);
    }
    const float b1v = b1[j * 16 + lm];
    const float w2v = W2[j * 16 + lm];
#pragma unroll
    for (int r = 0; r < 8; ++r)
      sacc[r] += tanh_hw(c[r] + b1v) * w2v;
  }
#pragma unroll
  for (int r = 0; r < 8; ++r) {   // reduce over N within each 16-lane half
    float s = sacc[r];
    s += __shfl_xor(s, 1);
    s += __shfl_xor(s, 2);
    s += __shfl_xor(s, 4);
    s += __shfl_xor(s, 8);
    sacc[r] = s;
  }
  if (lm == 0) {
    const float bb = b2[0];
#pragma unroll
    for (int r = 0; r < 8; ++r)
      scores[wave * 16 + half * 8 + r] = sacc[r] + bb;
  }
}

// One wave per team: segment softmax + weighted row-sum. Writes pre-projection
// team_h into d_out head and attn weights into d_out tail.
__global__ __launch_bounds__(256) void team_kernel(
    const float* __restrict__ agent_h, const float* __restrict__ scores,
    const int* __restrict__ offsets, const int* __restrict__ bucket,
    float* __restrict__ team_out, float* __restrict__ attn_out) {
  const int t    = blockIdx.x * 8 + (threadIdx.x >> 5);
  const int lane = threadIdx.x & 31;
  const int beg  = offsets[t];
  const int end  = offsets[t + 1];

  float acc[8] = {0.f, 0.f, 0.f, 0.f, 0.f, 0.f, 0.f, 0.f};
  if (end > beg) {
    float mx = -3.402823466e38f;
    for (int i = beg + lane; i < end; i += 32)
      mx = fmaxf(mx, scores[bucket[i]]);
    mx = fmaxf(mx, __shfl_xor(mx, 1));
    mx = fmaxf(mx, __shfl_xor(mx, 2));
    mx = fmaxf(mx, __shfl_xor(mx, 4));
    mx = fmaxf(mx, __shfl_xor(mx, 8));
    mx = fmaxf(mx, __shfl_xor(mx, 16));

    float den = 0.f;
    for (int i = beg + lane; i < end; i += 32)
      den += __expf(scores[bucket[i]] - mx);
    den += __shfl_xor(den, 1);
    den += __shfl_xor(den, 2);
    den += __shfl_xor(den, 4);
    den += __shfl_xor(den, 8);
    den += __shfl_xor(den, 16);
    const float inv = 1.0f / den;

    for (int i = beg; i < end; ++i) {
      const int ag  = bucket[i];
      const float w = __expf(scores[ag] - mx) * inv;
      if (lane == 0) attn_out[ag] = w;
      const float* row = agent_h + (size_t)ag * HIDDEN + lane * 8;
      const float4 r0 = *(const float4*)(row);
      const float4 r1 = *(const float4*)(row + 4);
      acc[0] += w * r0.x; acc[1] += w * r0.y; acc[2] += w * r0.z; acc[3] += w * r0.w;
      acc[4] += w * r1.x; acc[5] += w * r1.y; acc[6] += w * r1.z; acc[7] += w * r1.w;
    }
  }
  float* orow = team_out + (size_t)t * HIDDEN + lane * 8;
  *(float4*)(orow)     = make_float4(acc[0], acc[1], acc[2], acc[3]);
  *(float4*)(orow + 4) = make_float4(acc[4], acc[5], acc[6], acc[7]);
}

// In-place relu(team_h @ Wo + bo) on d_out. Each wave owns 16 rows and loads
// all 16x256 inputs to registers before any store, so in-place is race-free.
// Wo stays on the global path (L2-resident) to preserve row ownership.
__global__ __launch_bounds__(256) void proj_kernel(
    const float* __restrict__ Wo, const float* __restrict__ bo,
    float* __restrict__ team_io) {
  const int wave = blockIdx.x * 8 + (threadIdx.x >> 5);
  if (wave >= NTM / 16) return;
  const int lane = threadIdx.x & 31;
  const int half = lane >> 4;
  const int lm   = lane & 15;

  const float* arow = team_io + (size_t)(wave * 16 + lm) * HIDDEN + half * 2;
  v2f a[64];
#pragma unroll
  for (int kk = 0; kk < 64; ++kk)
    a[kk] = *(const v2f*)(arow + kk * 4);

  for (int j = 0; j < 16; ++j) {
    v8f c = {0.f, 0.f, 0.f, 0.f, 0.f, 0.f, 0.f, 0.f};
    const float* bcol = Wo + (size_t)(half * 2) * HIDDEN + j * 16 + lm;
#pragma unroll
    for (int kk = 0; kk < 64; ++kk) {
      v2f b;
      b.x = bcol[(size_t)kk * 4 * HIDDEN];
      b.y = bcol[(size_t)kk * 4 * HIDDEN + HIDDEN];
      c = __builtin_amdgcn_wmma_f32_16x16x4_f32(false, a[kk], false, b,
                                                (short)0, c, false, false);
    }
    const float bov = bo[j * 16 + lm];
#pragma unroll
    for (int r = 0; r < 8; ++r) {
      float v = c[r] + bov;
      v = v > 0.f ? v : 0.f;
      team_io[(size_t)(wave * 16 + half * 8 + r) * HIDDEN + j * 16 + lm] = v;
    }
  }
}

extern "C" void kernel_launch(void* const* d_in, const int* in_sizes, int n_in,
                              void* d_out, int out_size, void* d_ws, size_t ws_size,
                              hipStream_t stream) {
  (void)in_sizes; (void)n_in; (void)out_size; (void)ws_size;
  const float* agent_h  = (const float*)d_in[0];
  const int*   team_idx = (const int*)d_in[1];   // jax default x64-off -> int32
  // d_in[2] = n_teams scalar (compile-time constant here)
  const float* W1 = (const float*)d_in[3];
  const float* b1 = (const float*)d_in[4];
  const float* W2 = (const float*)d_in[5];
  const float* b2 = (const float*)d_in[6];
  const float* Wo = (const float*)d_in[7];
  const float* bo = (const float*)d_in[8];

  float* team_out = (float*)d_out;                       // [NTM*HIDDEN]
  float* attn_out = team_out + (size_t)NTM * HIDDEN;     // [NAG]

  char* ws = (char*)d_ws;
  float* scores  = (float*)ws;  ws += sizeof(float) * NAG;
  int*   counts  = (int*)ws;    ws += sizeof(int) * NTM;
  int*   offsets = (int*)ws;    ws += sizeof(int) * (NTM + 1);
  int*   cursor  = (int*)ws;    ws += sizeof(int) * NTM;
  int*   bucket  = (int*)ws;    // NAG ints; total ws use ~4.4 MB

  zero_kernel<<<(NTM + 255) / 256, 256, 0, stream>>>(counts, NTM);
  count_kernel<<<(NAG + 255) / 256, 256, 0, stream>>>(team_idx, counts);
  scan_kernel<<<1, 1024, 0, stream>>>(counts, offsets, cursor);
  scatter_kernel<<<(NAG + 255) / 256, 256, 0, stream>>>(team_idx, cursor, bucket);
  score_kernel<<<NAG / 16 / 8, 256, 0, stream>>>(agent_h, W1, b1, W2, b2, scores);
  team_kernel<<<NTM / 8, 256, 0, stream>>>(agent_h, scores, offsets, bucket,
                                           team_out, attn_out);
  proj_kernel<<<(NTM / 16 + 7) / 8, 256, 0, stream>>>(Wo, bo, team_out);
}